// SAGE_VE_29755533426986
// MI455X (gfx1250) — compile-verified
//
#include <hip/hip_runtime.h>
#include <math.h>

// Problem constants (from the reference)
#define N_NODES 200000
#define N_EDGES 3200000
#define IN_DIM  16
#define EMB_DIM 8

#define TILE 16
#define N_TILES (N_NODES / TILE)      // 12500 exact (200000 = 16*12500)
#define WAVES_PER_BLOCK 8             // 256 threads = 8 wave32

typedef __attribute__((ext_vector_type(2))) float v2f;
typedef __attribute__((ext_vector_type(8))) float v8f;

// ---------------------------------------------------------------------------
// Zero the accumulation regions of the workspace.
// ---------------------------------------------------------------------------
__global__ void sage_zero(float* __restrict__ p, int n) {
    int i = blockIdx.x * blockDim.x + threadIdx.x;
    int stride = gridDim.x * blockDim.x;
    for (; i < n; i += stride) p[i] = 0.0f;
}

// ---------------------------------------------------------------------------
// Pass 1 edge scatter, (edge, dim) mapping: 16 consecutive lanes cooperate on
// one edge. x-row gather is a coalesced 64B read; the 16 f32 atomics per edge
// hit a single agg cacheline from adjacent lanes (L2 merges them); src/dst/w
// loads are lane-broadcast cache hits. Lane with d==0 bumps the degree count.
// ---------------------------------------------------------------------------
__global__ void sage_edge_scatter16(const float* __restrict__ x,
                                    const int*   __restrict__ ei,
                                    const float* __restrict__ ew,
                                    float* __restrict__ agg,
                                    float* __restrict__ cnt) {
    const int total = N_EDGES * IN_DIM;   // 51.2M, fits in int
    int i = blockIdx.x * blockDim.x + threadIdx.x;
    int stride = gridDim.x * blockDim.x;
    for (; i < total; i += stride) {
        int e = i >> 4;
        int d = i & 15;
        int   src = ei[e];
        int   dst = ei[N_EDGES + e];
        float w   = ew[e];
        float v   = x[(size_t)src * IN_DIM + d] * w;
        atomicAdd(agg + (size_t)dst * IN_DIM + d, v);
        if (d == 0) atomicAdd(cnt + dst, 1.0f);
    }
}

// ---------------------------------------------------------------------------
// Pass 2 edge scatter (8-dim hidden): 8 lanes per edge. Counts reused.
// ---------------------------------------------------------------------------
__global__ void sage_edge_scatter8(const float* __restrict__ h,
                                   const int*   __restrict__ ei,
                                   const float* __restrict__ ew,
                                   float* __restrict__ agg) {
    const int total = N_EDGES * EMB_DIM;  // 25.6M
    int i = blockIdx.x * blockDim.x + threadIdx.x;
    int stride = gridDim.x * blockDim.x;
    for (; i < total; i += stride) {
        int e = i >> 3;
        int d = i & 7;
        int   src = ei[e];
        int   dst = ei[N_EDGES + e];
        float w   = ew[e];
        float v   = h[(size_t)src * EMB_DIM + d] * w;
        atomicAdd(agg + (size_t)dst * EMB_DIM + d, v);
    }
}

// ---------------------------------------------------------------------------
// Layer 1 transform (WMMA f32 16x16x4):
//   h = relu( (agg1/max(cnt,1)) @ W1_l.T + x @ W1_r.T + b1 )
// One wave per 16-node tile. B[k][n] = W[n][k], N padded 8->16 with a 0/1
// mask (branchless: row index clamped with n&7 so every load is valid).
// A layout per ISA: lane<16 -> K = 4kb+{0,1}; lane>=16 -> K = 4kb+{2,3}.
// D layout: VGPR v, lane L == D[v + 8*(L>>4)][L&15] -> restage via LDS.
// ---------------------------------------------------------------------------
__global__ void __launch_bounds__(256)
sage_layer1(const float* __restrict__ x,
            const float* __restrict__ agg,
            const float* __restrict__ cnt,
            const float* __restrict__ Wl,   // [8][16]
            const float* __restrict__ Wr,   // [8][16]
            const float* __restrict__ bias, // [8]
            float* __restrict__ h) {        // [N][8]
    __shared__ float smem[WAVES_PER_BLOCK][TILE][TILE];

    int lane = threadIdx.x & 31;
    int wave = threadIdx.x >> 5;
    int tile = blockIdx.x * WAVES_PER_BLOCK + wave;
    bool active = (tile < N_TILES);
    int tc = active ? tile : (N_TILES - 1);   // clamp: keep EXEC all-1s for WMMA

    int row  = lane & 15;      // M for A-loads, N for B-loads
    int half = lane >> 4;      // K half selector
    int node = tc * TILE + row;
    int n    = row;            // B column this lane supplies
    int nw   = n & 7;          // clamped weight row (always valid)
    float msk = (n < EMB_DIM) ? 1.0f : 0.0f;

    float inv = __builtin_amdgcn_rcpf(fmaxf(cnt[node], 1.0f));
    const float* arow  = agg + (size_t)node * IN_DIM;
    const float* xrow  = x   + (size_t)node * IN_DIM;
    const float* wlrow = Wl  + (size_t)nw * IN_DIM;
    const float* wrrow = Wr  + (size_t)nw * IN_DIM;

    v8f acc = {0.f, 0.f, 0.f, 0.f, 0.f, 0.f, 0.f, 0.f};

#pragma unroll
    for (int kb = 0; kb < 4; ++kb) {           // mean @ W1_l.T
        int koff = kb * 4 + half * 2;
        v2f a;  a.x = arow[koff] * inv;    a.y = arow[koff + 1] * inv;
        v2f b;  b.x = wlrow[koff] * msk;   b.y = wlrow[koff + 1] * msk;
        acc = __builtin_amdgcn_wmma_f32_16x16x4_f32(false, a, false, b,
                                                    (short)0, acc, false, false);
    }
#pragma unroll
    for (int kb = 0; kb < 4; ++kb) {           // + x @ W1_r.T
        int koff = kb * 4 + half * 2;
        v2f a;  a.x = xrow[koff];          a.y = xrow[koff + 1];
        v2f b;  b.x = wrrow[koff] * msk;   b.y = wrrow[koff + 1] * msk;
        acc = __builtin_amdgcn_wmma_f32_16x16x4_f32(false, a, false, b,
                                                    (short)0, acc, false, false);
    }

    float bv = bias[nw] * msk;
#pragma unroll
    for (int v = 0; v < 8; ++v) {
        int m = v + half * 8;
        smem[wave][m][n] = fmaxf(acc[v] + bv, 0.0f);   // bias + relu
    }
    __syncthreads();

    if (active) {
        // lane L writes 4 floats of node row (L>>1), half g = L&1
        int r = lane >> 1, g = lane & 1;
        int onode = tile * TILE + r;
        float4 o;
        o.x = smem[wave][r][g * 4 + 0];
        o.y = smem[wave][r][g * 4 + 1];
        o.z = smem[wave][r][g * 4 + 2];
        o.w = smem[wave][r][g * 4 + 3];
        *(float4*)(h + (size_t)onode * EMB_DIM + g * 4) = o;
    }
}

// ---------------------------------------------------------------------------
// Layer 2 transform + log_softmax (WMMA f32 16x16x4, K=8 -> 2 slices/matrix):
//   z = (agg2/max(cnt,1)) @ W3_l.T + h @ W3_r.T + b3 ; out = log_softmax(z)
// ---------------------------------------------------------------------------
__global__ void __launch_bounds__(256)
sage_layer2(const float* __restrict__ h,    // [N][8]
            const float* __restrict__ agg,  // [N][8]
            const float* __restrict__ cnt,
            const float* __restrict__ Wl,   // [8][8]
            const float* __restrict__ Wr,   // [8][8]
            const float* __restrict__ bias, // [8]
            float* __restrict__ out) {      // [N][8]
    __shared__ float smem[WAVES_PER_BLOCK][TILE][TILE];

    int lane = threadIdx.x & 31;
    int wave = threadIdx.x >> 5;
    int tile = blockIdx.x * WAVES_PER_BLOCK + wave;
    bool active = (tile < N_TILES);
    int tc = active ? tile : (N_TILES - 1);

    int row  = lane & 15;
    int half = lane >> 4;
    int node = tc * TILE + row;
    int n    = row;
    int nw   = n & 7;
    float msk = (n < EMB_DIM) ? 1.0f : 0.0f;

    float inv = __builtin_amdgcn_rcpf(fmaxf(cnt[node], 1.0f));
    const float* arow  = agg + (size_t)node * EMB_DIM;
    const float* hrow  = h   + (size_t)node * EMB_DIM;
    const float* wlrow = Wl  + (size_t)nw * EMB_DIM;
    const float* wrrow = Wr  + (size_t)nw * EMB_DIM;

    v8f acc = {0.f, 0.f, 0.f, 0.f, 0.f, 0.f, 0.f, 0.f};

#pragma unroll
    for (int kb = 0; kb < 2; ++kb) {           // mean2 @ W3_l.T  (K = 0..7)
        int koff = kb * 4 + half * 2;
        v2f a;  a.x = arow[koff] * inv;    a.y = arow[koff + 1] * inv;
        v2f b;  b.x = wlrow[koff] * msk;   b.y = wlrow[koff + 1] * msk;
        acc = __builtin_amdgcn_wmma_f32_16x16x4_f32(false, a, false, b,
                                                    (short)0, acc, false, false);
    }
#pragma unroll
    for (int kb = 0; kb < 2; ++kb) {           // + h @ W3_r.T
        int koff = kb * 4 + half * 2;
        v2f a;  a.x = hrow[koff];          a.y = hrow[koff + 1];
        v2f b;  b.x = wrrow[koff] * msk;   b.y = wrrow[koff + 1] * msk;
        acc = __builtin_amdgcn_wmma_f32_16x16x4_f32(false, a, false, b,
                                                    (short)0, acc, false, false);
    }

    float bv = bias[nw] * msk;
#pragma unroll
    for (int v = 0; v < 8; ++v) {
        int m = v + half * 8;
        smem[wave][m][n] = acc[v] + bv;
    }
    __syncthreads();

    if (active && lane < TILE) {
        int onode = tile * TILE + lane;
        float z[EMB_DIM];
        float mx = -INFINITY;
#pragma unroll
        for (int j = 0; j < EMB_DIM; ++j) { z[j] = smem[wave][lane][j]; mx = fmaxf(mx, z[j]); }
        float s = 0.0f;
#pragma unroll
        for (int j = 0; j < EMB_DIM; ++j) s += expf(z[j] - mx);
        float lse = mx + logf(s);
        float4 o0, o1;
        o0.x = z[0] - lse; o0.y = z[1] - lse; o0.z = z[2] - lse; o0.w = z[3] - lse;
        o1.x = z[4] - lse; o1.y = z[5] - lse; o1.z = z[6] - lse; o1.w = z[7] - lse;
        float* orow = out + (size_t)onode * EMB_DIM;
        *(float4*)(orow + 0) = o0;
        *(float4*)(orow + 4) = o1;
    }
}

// ---------------------------------------------------------------------------
// Host launcher. Workspace layout (floats):
//   [0)              agg1 : N*16
//   [N*16)           cnt  : N
//   [N*17)           agg2 : N*8
//   [N*25)           h    : N*8
// Total = N*33*4 bytes = 26.4 MB.
// ---------------------------------------------------------------------------
extern "C" void kernel_launch(void* const* d_in, const int* in_sizes, int n_in,
                              void* d_out, int out_size, void* d_ws, size_t ws_size,
                              hipStream_t stream) {
    const float* x    = (const float*)d_in[0];
    const int*   ei   = (const int*)  d_in[1];
    const float* ew   = (const float*)d_in[2];
    const float* W1_l = (const float*)d_in[3];
    const float* W1_r = (const float*)d_in[4];
    const float* b1   = (const float*)d_in[5];
    const float* W3_l = (const float*)d_in[6];
    const float* W3_r = (const float*)d_in[7];
    const float* b3   = (const float*)d_in[8];
    float* out = (float*)d_out;

    float* ws   = (float*)d_ws;
    float* agg1 = ws;                          // N*16
    float* cnt  = ws + (size_t)N_NODES * 16;   // N
    float* agg2 = cnt + N_NODES;               // N*8
    float* h    = agg2 + (size_t)N_NODES * 8;  // N*8

    // 1) zero accumulators (agg1 + cnt + agg2 = N*25 floats; h fully overwritten)
    int nzero = N_NODES * 25;
    sage_zero<<<2048, 256, 0, stream>>>(ws, nzero);

    // 2) edge scatter pass 1 (16-dim) + degree counts
    int e16blocks = (N_EDGES * IN_DIM + 255) / 256;
    sage_edge_scatter16<<<e16blocks, 256, 0, stream>>>(x, ei, ew, agg1, cnt);

    // 3) layer-1 transform (WMMA) + relu
    int lblocks = (N_TILES + WAVES_PER_BLOCK - 1) / WAVES_PER_BLOCK;
    sage_layer1<<<lblocks, 256, 0, stream>>>(x, agg1, cnt, W1_l, W1_r, b1, h);

    // 4) edge scatter pass 2 (8-dim), reusing counts
    int e8blocks = (N_EDGES * EMB_DIM + 255) / 256;
    sage_edge_scatter8<<<e8blocks, 256, 0, stream>>>(h, ei, ew, agg2);

    // 5) layer-2 transform (WMMA) + bias + log_softmax
    sage_layer2<<<lblocks, 256, 0, stream>>>(h, agg2, cnt, W3_l, W3_r, b3, out);
}